// GCNConv_28303834481380
// MI455X (gfx1250) — compile-verified
//
#include <hip/hip_runtime.h>
#include <cmath>

typedef __attribute__((ext_vector_type(2))) float v2f;
typedef __attribute__((ext_vector_type(8))) float v8f;

#define WT_STRIDE 66  // pad 64-float rows to 66 -> conflict-free ds_load_b64 across 16 lanes

// ---------------------------------------------------------------------------
// 1) zero output accumulator + degree buffer
__global__ __launch_bounds__(256) void gcn_zero(float* __restrict__ out,
                                                float* __restrict__ deg,
                                                int total, int n) {
  int i = blockIdx.x * 256 + threadIdx.x;
  if (i < total) out[i] = 0.0f;
  if (i < n) deg[i] = 0.0f;
}

// 2) degree over dst (self loops added later as +1)
__global__ __launch_bounds__(256) void gcn_degree(const int* __restrict__ dst,
                                                  float* __restrict__ deg, int E) {
  int e = blockIdx.x * 256 + threadIdx.x;
  if (e < E) unsafeAtomicAdd(&deg[dst[e]], 1.0f);
}

// 3) inv_sqrt = (deg + 1)^-1/2   (deg >= 1 always, no clipping needed)
__global__ __launch_bounds__(256) void gcn_invsqrt(const float* __restrict__ deg,
                                                   float* __restrict__ inv, int n) {
  int i = blockIdx.x * 256 + threadIdx.x;
  if (i < n) inv[i] = rsqrtf(deg[i] + 1.0f);
}

// ---------------------------------------------------------------------------
// 4) h = x @ W + bias, fp32 WMMA 16x16x4. One wave owns a 16x64 output tile.
// A (16x4): lane(m = lane%16, g = lane/16): a[i] = x[row0+m][kb + 2g + i]
// B (4x16): b[i] = W[kb + 2g + i][n0 + m]  == Wt[n0+m][kb + 2g + i] (LDS, transposed)
// C/D (16x16): acc[r] -> h[row0 + r + 8g][n0 + m]
__global__ __launch_bounds__(256) void gcn_gemm(const float* __restrict__ x,
                                                const float* __restrict__ w,
                                                const float* __restrict__ bias,
                                                float* __restrict__ h, int n) {
  __shared__ float wt[64 * WT_STRIDE];
  const int t = threadIdx.x;
  // stage W transposed: wt[o][k] = w[k][o]
  for (int idx = t; idx < 64 * 64; idx += 256) {
    int k = idx >> 6;
    int o = idx & 63;
    wt[o * WT_STRIDE + k] = w[idx];
  }
  __syncthreads();

  const int lane = t & 31;
  const int wv = t >> 5;
  const int m = lane & 15;
  const int g = lane >> 4;
  const long row0 = ((long)blockIdx.x * 8 + wv) * 16;
  if (row0 >= n) return;  // wave-uniform: EXEC stays all-ones for WMMA

  v8f acc[4];
#pragma unroll
  for (int nt = 0; nt < 4; ++nt) {
    float bv = bias[nt * 16 + m];
    acc[nt] = (v8f){bv, bv, bv, bv, bv, bv, bv, bv};
  }

  const float* xrow = x + (row0 + m) * 64;
#pragma unroll
  for (int kb = 0; kb < 64; kb += 4) {
    v2f a = *(const v2f*)(xrow + kb + 2 * g);
#pragma unroll
    for (int nt = 0; nt < 4; ++nt) {
      v2f b = *(const v2f*)(&wt[(nt * 16 + m) * WT_STRIDE + kb + 2 * g]);
      acc[nt] = __builtin_amdgcn_wmma_f32_16x16x4_f32(
          /*neg_a=*/false, a, /*neg_b=*/false, b,
          /*c_mod=*/(short)0, acc[nt], /*reuse_a=*/false, /*reuse_b=*/false);
    }
  }

#pragma unroll
  for (int nt = 0; nt < 4; ++nt) {
#pragma unroll
    for (int r = 0; r < 8; ++r) {
      h[(row0 + r + 8 * g) * 64 + nt * 16 + m] = acc[nt][r];
    }
  }
}

// ---------------------------------------------------------------------------
// 5) edge scatter: one wave per edge, 2 features per lane, fp32 HW atomics (L2)
__global__ __launch_bounds__(256) void gcn_scatter(const int* __restrict__ src,
                                                   const int* __restrict__ dst,
                                                   const float* __restrict__ h,
                                                   const float* __restrict__ inv,
                                                   float* __restrict__ out, int E) {
  const int lane = threadIdx.x & 31;
  const int e = blockIdx.x * 8 + (threadIdx.x >> 5);
  if (e >= E) return;
  const int s = src[e];
  const int d = dst[e];
  const float norm = inv[s] * inv[d];
  const float2 v = ((const float2*)(h + (size_t)s * 64))[lane];
  float* op = out + (size_t)d * 64 + lane * 2;
  unsafeAtomicAdd(op, v.x * norm);
  unsafeAtomicAdd(op + 1, v.y * norm);
}

// ---------------------------------------------------------------------------
// 6) out = gelu_tanh(out + h * inv^2)   (self-loop folded in, no atomics)
__global__ __launch_bounds__(256) void gcn_finalize(float* __restrict__ out,
                                                    const float* __restrict__ h,
                                                    const float* __restrict__ inv,
                                                    long total) {
  long i = (long)blockIdx.x * 256 + threadIdx.x;
  if (i >= total) return;
  const float iv = inv[i >> 6];
  const float v = out[i] + h[i] * iv * iv;
  const float c0 = 0.7978845608028654f;  // sqrt(2/pi)
  const float tnh = tanhf(c0 * (v + 0.044715f * v * v * v));
  out[i] = 0.5f * v * (1.0f + tnh);
}

// ---------------------------------------------------------------------------
extern "C" void kernel_launch(void* const* d_in, const int* in_sizes, int n_in,
                              void* d_out, int out_size, void* d_ws, size_t ws_size,
                              hipStream_t stream) {
  const float* x    = (const float*)d_in[0];  // [N,64]
  const int*   edge = (const int*)d_in[1];    // [2,E]
  const float* w    = (const float*)d_in[2];  // [64,64]
  const float* bias = (const float*)d_in[3];  // [64]
  float* out = (float*)d_out;

  const int N = in_sizes[0] / 64;
  const int E = in_sizes[1] / 2;
  const int* srcA = edge;
  const int* dstA = edge + E;

  char* ws = (char*)d_ws;
  float* h   = (float*)ws;                              // N*64 floats (25.6 MB)
  float* inv = (float*)(ws + (size_t)N * 64 * sizeof(float));
  float* deg = inv + N;

  const long total = (long)N * 64;

  gcn_zero<<<(int)((total + 255) / 256), 256, 0, stream>>>(out, deg, (int)total, N);
  gcn_degree<<<(E + 255) / 256, 256, 0, stream>>>(dstA, deg, E);
  gcn_invsqrt<<<(N + 255) / 256, 256, 0, stream>>>(deg, inv, N);
  gcn_gemm<<<(N + 127) / 128, 256, 0, stream>>>(x, w, bias, h, N);
  gcn_scatter<<<(E + 7) / 8, 256, 0, stream>>>(srcA, dstA, h, inv, out, E);
  gcn_finalize<<<(int)((total + 255) / 256), 256, 0, stream>>>(out, h, inv, total);
}